// RationalFourierOperator3D_8358006358667
// MI455X (gfx1250) — compile-verified
//
#include <hip/hip_runtime.h>
#include <math.h>

// ---------------------------------------------------------------------------
// RationalFourierOperator3D on MI455X (gfx1250, wave32, WMMA).
//
// All Fourier transforms are truncated DFTs expressed as dense GEMMs and run
// on V_WMMA_F32_16X16X4_F32 (fp32 matrix pipe, keeps reference precision).
// The dominant channel-mix GEMM stages its activation panel in LDS (34 KB)
// so each field element is read from HBM once instead of 4x.
// Required d_ws: ~396.5 MB (99,120,640 floats).
// ---------------------------------------------------------------------------

typedef __attribute__((ext_vector_type(2))) float v2f;
typedef __attribute__((ext_vector_type(8))) float v8f;

#define DEV static __device__ __forceinline__

DEV v8f wmma4(v2f a, v2f b, v8f c) {
  // D = A(16x4 f32) * B(4x16 f32) + C(16x16 f32)
  return __builtin_amdgcn_wmma_f32_16x16x4_f32(false, a, false, b, (short)0, c,
                                               false, false);
}

DEV float ipowf(float x, int p) {
  float r = 1.f;
  for (int i = 0; i < p; ++i) r *= x;
  return r;
}

// ---------------- constants ----------------
// grid: 64^3, modes 16x16x9, channels W=64, B=2, ORDER=4 -> 64 pqr terms
#define NS 262144            // 64^3 spatial points
#define K3 2304              // 16*16*9 retained modes
#define LPAD 68              // LDS row stride (floats) for the conv panel

// ---------------- table init ----------------
__global__ __launch_bounds__(256) void k_init(float* F1T, float* FYT, float* IT,
                                              float* I3T, float* BAS,
                                              float* DEC) {
  int idx = blockIdx.x * 256 + threadIdx.x;
  const float PI2 = 6.2831853071795864769f;
  if (idx < 64 * 32) {  // forward z: [z][2kz+c], cos / -sin
    int z = idx >> 5, col = idx & 31, kz = col >> 1;
    float v = 0.f;
    if (kz < 9) {
      float th = PI2 * (float)(kz * z) / 64.f;
      v = (col & 1) ? -__sinf(th) : __cosf(th);
    }
    F1T[idx] = v;
  }
  if (idx < 128 * 32) {  // forward y/x: [2y+c][2ky+d]
    int j = idx >> 5, col = idx & 31;
    int c = j & 1, y = j >> 1, ky = col >> 1, d = col & 1;
    float ph = PI2 * (float)(ky * y) / 64.f;
    float s = __sinf(ph), co = __cosf(ph);
    FYT[idx] = (d == 0) ? (c == 0 ? co : s) : (c == 0 ? -s : co);
  }
  if (idx < 32 * 128) {  // inverse x/y: [2k+c][2n+d]
    int j = idx >> 7, col = idx & 127;
    int k = j >> 1, c = j & 1, n = col >> 1, d = col & 1;
    float th = PI2 * (float)(k * n) / 64.f;
    float s = __sinf(th), co = __cosf(th);
    IT[idx] = (d == 0) ? (c == 0 ? co : -s) : (c == 0 ? s : co);
  }
  if (idx < 20 * 64) {  // inverse z (+hermitian weight, 1/64^3): [2kz+c][z]
    int j = idx >> 6, z = idx & 63;
    int kz = j >> 1, c = j & 1;
    float v = 0.f;
    if (kz < 9) {
      float w = ((kz == 0) ? 1.f : 2.f) / 262144.f;
      float th = PI2 * (float)(kz * z) / 64.f;
      v = (c == 0) ? w * __cosf(th) : -w * __sinf(th);
    }
    I3T[idx] = v;
  }
  if (idx < 64 * K3) {  // poly basis: [pqr][k3], masked p+q+r<4
    int pqr = idx / K3, k3 = idx % K3;
    int p = pqr >> 4, q = (pqr >> 2) & 3, r = pqr & 3;
    int kx = k3 / 144, ky = (k3 / 9) % 16, kz = k3 % 9;
    float v = 0.f;
    if (p + q + r < 4)
      v = ipowf(kx / 16.f, p) * ipowf(ky / 16.f, q) * ipowf(kz / 16.f, r);
    BAS[idx] = v;
  }
  if (idx < K3) {  // decay mask
    int kx = idx / 144, ky = (idx / 9) % 16, kz = idx % 9;
    float fx = kx / 16.f, fy = ky / 16.f, fz = kz / 16.f;
    DEC[idx] = __expf(-2.f * (fx * fx + fy * fy + fz * fz));
  }
}

// ---------------- input projection: h = x @ W_in + b_in ----------------
__global__ __launch_bounds__(256) void k_win(const float* __restrict__ x,
                                             const float* __restrict__ Win,
                                             const float* __restrict__ bin,
                                             float* __restrict__ h) {
  int gid = blockIdx.x * 256 + threadIdx.x;  // over 2*NS
  int b = gid / NS, s = gid % NS;
  float x0 = x[(b * 3 + 0) * NS + s];
  float x1 = x[(b * 3 + 1) * NS + s];
  float x2 = x[(b * 3 + 2) * NS + s];
  for (int w = 0; w < 64; ++w) {
    float v = x0 * Win[w] + x1 * Win[64 + w] + x2 * Win[128 + w] + bin[w];
    h[(b * 64 + w) * NS + s] = v;
  }
}

// ---------------- output projection ----------------
__global__ __launch_bounds__(256) void k_wout(const float* __restrict__ h,
                                              const float* __restrict__ Wout,
                                              const float* __restrict__ bout,
                                              float* __restrict__ out) {
  int gid = blockIdx.x * 256 + threadIdx.x;  // over 2*3*NS
  int b = gid / (3 * NS), o = (gid / NS) % 3, s = gid % NS;
  float acc = bout[o];
  for (int w = 0; w < 64; ++w) acc += h[(b * 64 + w) * NS + s] * Wout[w * 3 + o];
  out[gid] = acc;
}

// ---------------- Rk numerator/denominator GEMM: [4096x64]x[64x2304] --------
__global__ __launch_bounds__(256) void k_gemm_rk(const float* __restrict__ A,
                                                 const float* __restrict__ B,
                                                 float* __restrict__ C) {
  int gw = blockIdx.x * 8 + (threadIdx.x >> 5);  // 36864 tiles
  int lane = threadIdx.x & 31, half = lane >> 4, l16 = lane & 15;
  int mt = gw / 144, nt = gw % 144;
  int row0 = mt * 16, col0 = nt * 16;
  v8f acc = {0.f, 0.f, 0.f, 0.f, 0.f, 0.f, 0.f, 0.f};
  int ar = (row0 + l16) * 64, bc = col0 + l16;
  for (int k0 = 0; k0 < 64; k0 += 4) {
    int kk = k0 + 2 * half;
    v2f a, b;
    a.x = A[ar + kk];
    a.y = A[ar + kk + 1];
    b.x = B[kk * K3 + bc];
    b.y = B[(kk + 1) * K3 + bc];
    acc = wmma4(a, b, acc);
  }
#pragma unroll
  for (int v = 0; v < 8; ++v)
    C[(row0 + v + 8 * half) * K3 + col0 + l16] = acc[v];
}

__global__ __launch_bounds__(256) void k_rkfinal(float* __restrict__ Pk,
                                                 const float* __restrict__ Qk,
                                                 const float* __restrict__ DEC) {
  int gid = blockIdx.x * 256 + threadIdx.x;  // over 4096*2304
  int k3 = gid % K3;
  Pk[gid] = DEC[k3] * Pk[gid] / (Qk[gid] + 1e-6f);
}

// ---------------- forward z DFT: rows(b,i,x,y) K=z64 -> 18 cols -------------
__global__ __launch_bounds__(256) void k_f1(const float* __restrict__ h,
                                            const float* __restrict__ F1T,
                                            float* __restrict__ A1) {
  int gw = blockIdx.x * 8 + (threadIdx.x >> 5);  // 65536 tiles
  int lane = threadIdx.x & 31, half = lane >> 4, l16 = lane & 15;
  int mt = gw >> 1, nt = gw & 1;
  int row0 = mt * 16, col0 = nt * 16;
  v8f acc = {0.f, 0.f, 0.f, 0.f, 0.f, 0.f, 0.f, 0.f};
  int ar = (row0 + l16) * 64, bc = col0 + l16;
  for (int k0 = 0; k0 < 64; k0 += 4) {
    int kk = k0 + 2 * half;
    v2f a, b;
    a.x = h[ar + kk];
    a.y = h[ar + kk + 1];
    b.x = F1T[kk * 32 + bc];
    b.y = F1T[(kk + 1) * 32 + bc];
    acc = wmma4(a, b, acc);
  }
  int col = col0 + l16;
  if (col < 18) {
#pragma unroll
    for (int v = 0; v < 8; ++v) A1[(row0 + v + 8 * half) * 18 + col] = acc[v];
  }
}

// ---------------- forward y DFT: rows(b,i,x,kz) K=2y+c(128) -> 32 cols ------
__global__ __launch_bounds__(256) void k_f2(const float* __restrict__ A1,
                                            const float* __restrict__ FYT,
                                            float* __restrict__ A2) {
  int gw = blockIdx.x * 8 + (threadIdx.x >> 5);  // 9216 tiles
  int lane = threadIdx.x & 31, half = lane >> 4, l16 = lane & 15;
  int mt = gw >> 1, nt = gw & 1;
  int row0 = mt * 16, col0 = nt * 16;
  int r = row0 + l16, bix = r / 9, kz = r % 9;
  int abase = bix * 1152 + kz * 2;
  v8f acc = {0.f, 0.f, 0.f, 0.f, 0.f, 0.f, 0.f, 0.f};
  int bc = col0 + l16;
  for (int k0 = 0; k0 < 128; k0 += 4) {
    int kk = k0 + 2 * half, y = kk >> 1;
    v2f a, b;
    a.x = A1[abase + y * 18];
    a.y = A1[abase + y * 18 + 1];
    b.x = FYT[kk * 32 + bc];
    b.y = FYT[(kk + 1) * 32 + bc];
    acc = wmma4(a, b, acc);
  }
  int n = col0 + l16, ky = n >> 1, d = n & 1;
#pragma unroll
  for (int v = 0; v < 8; ++v) {
    int rr = row0 + v + 8 * half, bx = rr / 9, kz2 = rr % 9;
    A2[bx * 288 + ky * 18 + kz2 * 2 + d] = acc[v];
  }
}

// ---------------- forward x DFT: rows(b,i,ky,kz) K=2x+c -> x_modes ----------
__global__ __launch_bounds__(256) void k_f3(const float* __restrict__ A2,
                                            const float* __restrict__ FYT,
                                            float* __restrict__ XM) {
  int gw = blockIdx.x * 8 + (threadIdx.x >> 5);  // 2304 tiles
  int lane = threadIdx.x & 31, half = lane >> 4, l16 = lane & 15;
  int mt = gw >> 1, nt = gw & 1;
  int row0 = mt * 16, col0 = nt * 16;
  int r = row0 + l16, kz = r % 9, t = r / 9, ky = t & 15, bi = t >> 4;
  int abase = bi * 18432 + ky * 18 + kz * 2;
  v8f acc = {0.f, 0.f, 0.f, 0.f, 0.f, 0.f, 0.f, 0.f};
  int bc = col0 + l16;
  for (int k0 = 0; k0 < 128; k0 += 4) {
    int kk = k0 + 2 * half, x = kk >> 1;
    v2f a, b;
    a.x = A2[abase + x * 288];
    a.y = A2[abase + x * 288 + 1];
    b.x = FYT[kk * 32 + bc];
    b.y = FYT[(kk + 1) * 32 + bc];
    acc = wmma4(a, b, acc);
  }
  int n = col0 + l16, kx = n >> 1, d = n & 1;
#pragma unroll
  for (int v = 0; v < 8; ++v) {
    int rr = row0 + v + 8 * half, kz2 = rr % 9, t2 = rr / 9;
    int ky2 = t2 & 15, bi2 = t2 >> 4;
    XM[bi2 * 4608 + kx * 288 + ky2 * 18 + kz2 * 2 + d] = acc[v];
  }
}

// ---------------- per-mode rational mixing (real Rk, complex modes) ---------
__global__ __launch_bounds__(256) void k_mix(const float* __restrict__ XM,
                                             const float* __restrict__ Rk,
                                             float* __restrict__ OM) {
  int gid = blockIdx.x * 256 + threadIdx.x;  // over 2*64*2304*2
  int c = gid & 1, t = gid >> 1;
  int k3 = t % K3, t2 = t / K3, o = t2 & 63, b = t2 >> 6;
  int xbase = b * 294912 + k3 * 2 + c;  // i stride 4608
  int rbase = o * 147456 + k3;          // i stride 2304
  float acc = 0.f;
  for (int i = 0; i < 64; ++i) acc += XM[xbase + i * 4608] * Rk[rbase + i * K3];
  OM[(b * 64 + o) * 4608 + k3 * 2 + c] = acc;
}

// ---------------- inverse x: rows(b,o,ky,kz) K=2kx+c(32) -> (x,c)128 --------
__global__ __launch_bounds__(256) void k_i1(const float* __restrict__ OM,
                                            const float* __restrict__ IT,
                                            float* __restrict__ B1) {
  int gw = blockIdx.x * 8 + (threadIdx.x >> 5);  // 9216 tiles
  int lane = threadIdx.x & 31, half = lane >> 4, l16 = lane & 15;
  int mt = gw >> 3, nt = gw & 7;
  int row0 = mt * 16, col0 = nt * 16;
  int r = row0 + l16, kz = r % 9, t = r / 9, ky = t & 15, boi = t >> 4;
  int abase = boi * 4608 + ky * 18 + kz * 2;
  v8f acc = {0.f, 0.f, 0.f, 0.f, 0.f, 0.f, 0.f, 0.f};
  int bc = col0 + l16;
  for (int k0 = 0; k0 < 32; k0 += 4) {
    int kk = k0 + 2 * half, kx = kk >> 1;
    v2f a, b;
    a.x = OM[abase + kx * 288];
    a.y = OM[abase + kx * 288 + 1];
    b.x = IT[kk * 128 + bc];
    b.y = IT[(kk + 1) * 128 + bc];
    acc = wmma4(a, b, acc);
  }
  int n = col0 + l16, x = n >> 1, d = n & 1;
#pragma unroll
  for (int v = 0; v < 8; ++v) {
    int rr = row0 + v + 8 * half, kz2 = rr % 9, t2 = rr / 9;
    int ky2 = t2 & 15, boi2 = t2 >> 4;
    B1[boi2 * 18432 + x * 288 + ky2 * 18 + kz2 * 2 + d] = acc[v];
  }
}

// ---------------- inverse y: rows(b,o,x,kz) K=2ky+c -> (y,c)128 -------------
__global__ __launch_bounds__(256) void k_i2(const float* __restrict__ B1,
                                            const float* __restrict__ IT,
                                            float* __restrict__ B2) {
  int gw = blockIdx.x * 8 + (threadIdx.x >> 5);  // 36864 tiles
  int lane = threadIdx.x & 31, half = lane >> 4, l16 = lane & 15;
  int mt = gw >> 3, nt = gw & 7;
  int row0 = mt * 16, col0 = nt * 16;
  int r = row0 + l16, kz = r % 9, t = r / 9, x = t & 63, boi = t >> 6;
  int abase = boi * 18432 + x * 288 + kz * 2;
  v8f acc = {0.f, 0.f, 0.f, 0.f, 0.f, 0.f, 0.f, 0.f};
  int bc = col0 + l16;
  for (int k0 = 0; k0 < 32; k0 += 4) {
    int kk = k0 + 2 * half, ky = kk >> 1;
    v2f a, b;
    a.x = B1[abase + ky * 18];
    a.y = B1[abase + ky * 18 + 1];
    b.x = IT[kk * 128 + bc];
    b.y = IT[(kk + 1) * 128 + bc];
    acc = wmma4(a, b, acc);
  }
  int n = col0 + l16, y = n >> 1, d = n & 1;
#pragma unroll
  for (int v = 0; v < 8; ++v) {
    int rr = row0 + v + 8 * half, kz2 = rr % 9, t2 = rr / 9;
    int x2 = t2 & 63, boi2 = t2 >> 6;
    B2[(boi2 * 64 + x2) * 1152 + y * 18 + kz2 * 2 + d] = acc[v];
  }
}

// ---------------- inverse z + real part: rows(b,o,x,y) K=18(pad20) -> z64 ---
__global__ __launch_bounds__(256) void k_i3(const float* __restrict__ B2,
                                            const float* __restrict__ I3T,
                                            float* __restrict__ hf) {
  int gw = blockIdx.x * 8 + (threadIdx.x >> 5);  // 131072 tiles
  int lane = threadIdx.x & 31, half = lane >> 4, l16 = lane & 15;
  int mt = gw >> 2, nt = gw & 3;
  int row0 = mt * 16, col0 = nt * 16;
  int abase = (row0 + l16) * 18;
  v8f acc = {0.f, 0.f, 0.f, 0.f, 0.f, 0.f, 0.f, 0.f};
  int bc = col0 + l16;
  for (int k0 = 0; k0 < 20; k0 += 4) {
    int kk = k0 + 2 * half;
    v2f a, b;
    if (kk >= 18) {
      a.x = 0.f;
      a.y = 0.f;
    } else {
      a.x = B2[abase + kk];
      a.y = B2[abase + kk + 1];
    }
    b.x = I3T[kk * 64 + bc];
    b.y = I3T[(kk + 1) * 64 + bc];
    acc = wmma4(a, b, acc);
  }
#pragma unroll
  for (int v = 0; v < 8; ++v) hf[(row0 + v + 8 * half) * 64 + bc] = acc[v];
}

// ---- channel mix + bias + fourier + skip + GELU: h' = gelu(cw.h+cb+hf+h) ---
// One block = one batch b and one 128-wide spatial panel; the 64x128 activation
// panel is staged once in LDS (transposed, padded stride 68 -> conflict-free
// ds_load_b64 B-fragments), then 8 waves cover 4 o-tiles x 2 s-halves x 4
// N-subtiles. Cuts HBM reads of the field 4x vs per-tile global loads.
__global__ __launch_bounds__(256) void k_conv(const float* __restrict__ hA,
                                              const float* __restrict__ cw,
                                              const float* __restrict__ cb,
                                              float* __restrict__ hB) {
  __shared__ float smem[128 * LPAD];  // 34,816 B
  int tid = threadIdx.x;
  int b = blockIdx.x >> 11;           // 2048 panels per batch
  int s0 = (blockIdx.x & 2047) * 128;

  // cooperative fill: global [i][s] (coalesced) -> LDS [s][i]
  for (int idx = tid; idx < 64 * 128; idx += 256) {
    int i = idx >> 7, s = idx & 127;
    smem[s * LPAD + i] = hA[(b * 64 + i) * NS + s0 + s];
  }
  __syncthreads();

  int w = tid >> 5, lane = tid & 31, half = lane >> 4, l16 = lane & 15;
  int ot = w & 3, sh = w >> 2;        // o-tile, s-half
  int row0 = ot * 16;
  v8f acc0 = {0.f, 0.f, 0.f, 0.f, 0.f, 0.f, 0.f, 0.f};
  v8f acc1 = acc0, acc2 = acc0, acc3 = acc0;
  int ar = row0 + l16;
  int sbase = (sh * 64 + l16) * LPAD;
  for (int k0 = 0; k0 < 64; k0 += 4) {
    int kk = k0 + 2 * half;
    v2f a;
    a.x = cw[ar * 64 + kk];
    a.y = cw[ar * 64 + kk + 1];
    v2f b0, b1, b2, b3;
    b0.x = smem[sbase + kk];
    b0.y = smem[sbase + kk + 1];
    b1.x = smem[sbase + 16 * LPAD + kk];
    b1.y = smem[sbase + 16 * LPAD + kk + 1];
    b2.x = smem[sbase + 32 * LPAD + kk];
    b2.y = smem[sbase + 32 * LPAD + kk + 1];
    b3.x = smem[sbase + 48 * LPAD + kk];
    b3.y = smem[sbase + 48 * LPAD + kk + 1];
    acc0 = wmma4(a, b0, acc0);
    acc1 = wmma4(a, b1, acc1);
    acc2 = wmma4(a, b2, acc2);
    acc3 = wmma4(a, b3, acc3);
  }
#pragma unroll
  for (int n = 0; n < 4; ++n) {
    v8f acc = (n == 0) ? acc0 : (n == 1) ? acc1 : (n == 2) ? acc2 : acc3;
#pragma unroll
    for (int v = 0; v < 8; ++v) {
      int o = row0 + v + 8 * half;
      int idx = (b * 64 + o) * NS + s0 + sh * 64 + n * 16 + l16;
      float s = acc[v] + cb[o] + hB[idx] + hA[idx];
      hB[idx] = 0.5f * s * (1.f + erff(s * 0.70710678118654752f));
    }
  }
}

// ---------------------------------------------------------------------------
extern "C" void kernel_launch(void* const* d_in, const int* in_sizes, int n_in,
                              void* d_out, int out_size, void* d_ws,
                              size_t ws_size, hipStream_t stream) {
  const float* x = (const float*)d_in[0];
  const float* Win = (const float*)d_in[1];
  const float* bin = (const float*)d_in[2];
  const float* Wout = (const float*)d_in[3];
  const float* bout = (const float*)d_in[4];
  const float* P = (const float*)d_in[5];
  const float* Q = (const float*)d_in[6];
  const float* cw = (const float*)d_in[7];
  const float* cb = (const float*)d_in[8];
  float* out = (float*)d_out;
  float* ws = (float*)d_ws;

  // workspace layout (floats)
  float* bufA = ws;                      // 33554432 : h (ping)
  float* bufB = ws + 33554432ull;        // 33554432 : h (pong) / h_fourier
  float* A1 = ws + 67108864ull;          // 9437184  : fwd-z out, reused as B2
  float* A2 = ws + 76546048ull;          // 2359296  : fwd-y out, reused as B1
  float* XM = ws + 78905344ull;          // 589824   : x_modes
  float* OM = ws + 79495168ull;          // 589824   : out_modes
  float* Pk = ws + 80084992ull;          // 9437184  : Pk -> Rk
  float* Qk = ws + 89522176ull;          // 9437184  : Qk
  float* F1T = ws + 98959360ull;         // 2048
  float* FYT = ws + 98961408ull;         // 4096
  float* IT = ws + 98965504ull;          // 4096
  float* I3T = ws + 98969600ull;         // 1280
  float* BAS = ws + 98970880ull;         // 147456
  float* DEC = ws + 99118336ull;         // 2304   (total 99120640 floats)

  k_init<<<576, 256, 0, stream>>>(F1T, FYT, IT, I3T, BAS, DEC);
  k_win<<<2048, 256, 0, stream>>>(x, Win, bin, bufA);

  float* cur = bufA;
  float* nxt = bufB;
  for (int l = 0; l < 4; ++l) {
    const float* Pl = P + (size_t)l * 262144;
    const float* Ql = Q + (size_t)l * 262144;
    const float* cwl = cw + (size_t)l * 4096;
    const float* cbl = cb + (size_t)l * 64;
    // forward truncated DFT (z, then y, then x)
    k_f1<<<8192, 256, 0, stream>>>(cur, F1T, A1);
    k_f2<<<1152, 256, 0, stream>>>(A1, FYT, A2);
    k_f3<<<288, 256, 0, stream>>>(A2, FYT, XM);
    // rational transfer function Rk = decay * (P.basis)/(Q.basis+eps)
    k_gemm_rk<<<4608, 256, 0, stream>>>(Pl, BAS, Pk);
    k_gemm_rk<<<4608, 256, 0, stream>>>(Ql, BAS, Qk);
    k_rkfinal<<<36864, 256, 0, stream>>>(Pk, Qk, DEC);
    k_mix<<<2304, 256, 0, stream>>>(XM, Pk, OM);
    // inverse synthesis (x, then y, then z) -> h_fourier in nxt
    k_i1<<<1152, 256, 0, stream>>>(OM, IT, A2);   // A2 reused as B1
    k_i2<<<4608, 256, 0, stream>>>(A2, IT, A1);   // A1 reused as B2
    k_i3<<<16384, 256, 0, stream>>>(A1, I3T, nxt);
    // fused channel mix + bias + fourier + skip + GELU (LDS-staged panel)
    k_conv<<<4096, 256, 0, stream>>>(cur, cwl, cbl, nxt);
    float* t = cur;
    cur = nxt;
    nxt = t;
  }
  k_wout<<<6144, 256, 0, stream>>>(cur, Wout, bout, out);
}